// SSDMultiboxLoss_77378130804826
// MI455X (gfx1250) — compile-verified
//
#include <hip/hip_runtime.h>

// ---------------------------------------------------------------------------
// SSD multibox loss, MI455X (gfx1250).
// Memory-bound streaming reduction: ~151 MB read -> 1 float.
// Floor at 23.3 TB/s ~ 6.5 us. Strategy: max MLP, async-DMA the confs tile
// (9 planes x 256 anchors) into LDS via GLOBAL_LOAD_ASYNC_TO_LDS_B128 while
// the regression half streams through VGPRs; s_wait_asynccnt + barrier; then
// log-softmax/focal from LDS. wave32 shuffle reduction + 3 atomics per block.
// ---------------------------------------------------------------------------

constexpr int   kB            = 32;
constexpr int   kC            = 9;
constexpr int   kNA           = 65440;
constexpr int   kTile         = 256;
constexpr int   kTilesPerB    = (kNA + kTile - 1) / kTile;   // 256
constexpr float kScaleXY      = 10.0f;                       // 1/0.1
constexpr float kScaleWH      = 5.0f;                        // 1/0.2
constexpr float kAlphaSum     = 8010.0f;                     // 10 + 8*1000
constexpr float kInvMeanCount = 1.0f / (float)(kB * kNA);

#if __has_builtin(__builtin_amdgcn_global_load_async_to_lds_b128)
#define HAVE_ASYNC_LDS 1
typedef int v4i_t __attribute__((ext_vector_type(4)));
typedef __attribute__((address_space(1))) v4i_t* as1_v4i;
typedef __attribute__((address_space(3))) v4i_t* as3_v4i;
#else
#define HAVE_ASYNC_LDS 0
#endif

__device__ __forceinline__ void wait_async_lds() {
#if HAVE_ASYNC_LDS
#if __has_builtin(__builtin_amdgcn_s_wait_asynccnt)
  __builtin_amdgcn_s_wait_asynccnt(0);
#else
  asm volatile("s_wait_asynccnt 0" ::: "memory");
#endif
#endif
}

__device__ __forceinline__ float wave_sum(float v) {
#pragma unroll
  for (int off = 16; off > 0; off >>= 1) v += __shfl_down(v, off, 32);
  return v;
}

__global__ void __launch_bounds__(256)
ssd_loss_partial(const float* __restrict__ bbox_delta,   // (B,4,NA)
                 const float* __restrict__ confs,        // (B,C,NA)
                 const float* __restrict__ gt_bbox,      // (B,NA,4)
                 const int*   __restrict__ gt_labels,    // (B,NA)
                 const float* __restrict__ anchors,      // (1,4,NA)
                 float* __restrict__ ws)                  // [focal, reg, npos]
{
  __shared__ float sconf[kC * kTile];   // 9 KB confs tile, stride 256 -> no bank conflicts
  __shared__ float redF[8], redR[8], redP[8];

  const int tid     = threadIdx.x;
  const int b       = blockIdx.x / kTilesPerB;
  const int tileIdx = blockIdx.x % kTilesPerB;
  const int base    = tileIdx * kTile;
  const int a       = base + tid;
  const bool valid  = (a < kNA);
  const int  ac     = valid ? a : (kNA - 1);   // clamped, stays in-bounds

  // ---- 1) kick off async DMA of the confs tile: 9 planes x 1 KB -----------
  // 576 B128 transfers spread over 256 threads (9 planes * 64 quads).
  const float* cbase = confs + (size_t)b * kC * kNA;
#pragma unroll
  for (int k = 0; k < 3; ++k) {
    const int j = tid + (k << 8);
    if (j < kC * (kTile / 4)) {
      const int p   = j >> 6;          // plane 0..8
      const int q   = j & 63;          // 16B quad within tile row
      int col = base + (q << 2);
      if (col > kNA - 4) col = kNA - 4;            // tail clamp (invalid lanes only)
      float* src = const_cast<float*>(cbase + (size_t)p * kNA + col);
      float* dst = &sconf[p * kTile + (q << 2)];
#if HAVE_ASYNC_LDS
      __builtin_amdgcn_global_load_async_to_lds_b128((as1_v4i)src, (as3_v4i)dst,
                                                     /*offset=*/0, /*cpol=*/0);
#else
      *(float4*)dst = *(const float4*)src;          // fallback: sync staging
#endif
    }
  }

  // ---- 2) regression half (overlaps with the async copy) ------------------
  const float4 g  = *(const float4*)(gt_bbox + ((size_t)b * kNA + ac) * 4);
  const float  ax = anchors[0 * kNA + ac];
  const float  ay = anchors[1 * kNA + ac];
  const float  aw = anchors[2 * kNA + ac];
  const float  ah = anchors[3 * kNA + ac];
  const int    lbl = gt_labels[(size_t)b * kNA + ac];   // 0..8
  const float* bd  = bbox_delta + (size_t)b * 4 * kNA;
  const float  d0 = bd[0 * kNA + ac];
  const float  d1 = bd[1 * kNA + ac];
  const float  d2 = bd[2 * kNA + ac];
  const float  d3 = bd[3 * kNA + ac];

  float reg = 0.0f, pos = 0.0f;
  if (valid && lbl > 0) {
    const float t0 = kScaleXY * (g.x - ax) / aw;
    const float t1 = kScaleXY * (g.y - ay) / ah;
    const float t2 = kScaleWH * __logf(g.z / aw);
    const float t3 = kScaleWH * __logf(g.w / ah);
    float dd, adf;
    dd = d0 - t0; adf = fabsf(dd); reg += (adf < 1.0f) ? 0.5f * dd * dd : adf - 0.5f;
    dd = d1 - t1; adf = fabsf(dd); reg += (adf < 1.0f) ? 0.5f * dd * dd : adf - 0.5f;
    dd = d2 - t2; adf = fabsf(dd); reg += (adf < 1.0f) ? 0.5f * dd * dd : adf - 0.5f;
    dd = d3 - t3; adf = fabsf(dd); reg += (adf < 1.0f) ? 0.5f * dd * dd : adf - 0.5f;
    pos = 1.0f;
  }

  // ---- 3) classification half from LDS ------------------------------------
  wait_async_lds();     // issuing wave's ASYNCcnt -> 0
  __syncthreads();      // publish LDS tile to all 8 waves

  float focal = 0.0f;
  {
    float c[kC];
#pragma unroll
    for (int p = 0; p < kC; ++p) c[p] = sconf[p * kTile + tid];
    float m = c[0];
#pragma unroll
    for (int p = 1; p < kC; ++p) m = fmaxf(m, c[p]);
    float s = 0.0f;
#pragma unroll
    for (int p = 0; p < kC; ++p) s += __expf(c[p] - m);
    const float lse   = m + __logf(s);
    const float ct    = sconf[lbl * kTile + tid];   // dynamic class index via LDS
    const float logpt = ct - lse;
    const float pt    = __expf(logpt);
    const float om    = 1.0f - pt;
    if (valid) focal = om * om * om * logpt;        // (1-p)^3 * log p, GAMMA=3
  }

  // ---- 4) block reduction: wave32 shuffle tree + LDS + 3 atomics ----------
  const float fw = wave_sum(focal);
  const float rw = wave_sum(reg);
  const float pw = wave_sum(pos);
  const int wave = tid >> 5, lane = tid & 31;
  if (lane == 0) { redF[wave] = fw; redR[wave] = rw; redP[wave] = pw; }
  __syncthreads();
  if (tid == 0) {
    float sf = 0.0f, sr = 0.0f, sp = 0.0f;
#pragma unroll
    for (int w = 0; w < 8; ++w) { sf += redF[w]; sr += redR[w]; sp += redP[w]; }
    atomicAdd(ws + 0, sf);
    atomicAdd(ws + 1, sr);
    atomicAdd(ws + 2, sp);
  }
}

__global__ void ssd_loss_init(float* __restrict__ ws) {
  if (threadIdx.x < 3) ws[threadIdx.x] = 0.0f;
}

__global__ void ssd_loss_finalize(const float* __restrict__ ws,
                                  float* __restrict__ out) {
  if (threadIdx.x == 0) {
    const float focal_sum = ws[0];
    const float reg_sum   = ws[1];
    const float npos      = ws[2];
    out[0] = reg_sum / npos + (-kAlphaSum) * focal_sum * kInvMeanCount;
  }
}

extern "C" void kernel_launch(void* const* d_in, const int* in_sizes, int n_in,
                              void* d_out, int out_size, void* d_ws, size_t ws_size,
                              hipStream_t stream) {
  const float* bbox_delta = (const float*)d_in[0];
  const float* confs      = (const float*)d_in[1];
  const float* gt_bbox    = (const float*)d_in[2];
  const int*   gt_labels  = (const int*)d_in[3];
  const float* anchors    = (const float*)d_in[4];
  float* ws  = (float*)d_ws;
  float* out = (float*)d_out;

  ssd_loss_init<<<dim3(1), dim3(32), 0, stream>>>(ws);
  ssd_loss_partial<<<dim3(kB * kTilesPerB), dim3(kTile), 0, stream>>>(
      bbox_delta, confs, gt_bbox, gt_labels, anchors, ws);
  ssd_loss_finalize<<<dim3(1), dim3(32), 0, stream>>>(ws, out);
}